// Elbo_Loss_71004399337943
// MI455X (gfx1250) — compile-verified
//
#include <hip/hip_runtime.h>
#include <hip/hip_bf16.h>
#include <math.h>

#define MDIM 8192
#define KSEL 256
#define THRESHV 0.01f

typedef float v2f __attribute__((ext_vector_type(2)));
typedef float v8f __attribute__((ext_vector_type(8)));

// workspace layout (float offsets into d_ws)
#define WS_IND  0      // int[256]
#define WS_U    256    // float[256] top-k values
#define WS_W    512    // float[256] sqrt(vals)
#define WS_BZ   768    // float[256] w * z[ind]
#define WS_SCAL 1024   // [0]=kl, [1]=z.S, [2]=logdet
#define WS_BV   1056   // float[256] w * (Rk @ S)
#define WS_MSYM 2048   // float[256*256] symmetrized system, becomes L

// ---------------------------------------------------------------------------
// 1) top-256 by rank-selection: rank(i) = #{j : g[j] > g[i] or (==, j<i)}.
//    Strict total order -> ranks 0..255 each claimed exactly once; output
//    order matches jax.lax.top_k (descending, ties by index).
// ---------------------------------------------------------------------------
__global__ void topk_kernel(const float* __restrict__ gamma,
                            int* __restrict__ ind, float* __restrict__ u) {
  __shared__ float g[MDIM];
  for (int j = threadIdx.x; j < MDIM; j += blockDim.x) g[j] = gamma[j];
  __syncthreads();
  const int i = blockIdx.x * blockDim.x + threadIdx.x;
  const float gi = g[i];
  int rank = 0;
  for (int j = 0; j < MDIM; ++j) {
    const float gj = g[j];
    rank += (gj > gi || (gj == gi && j < i)) ? 1 : 0;
  }
  if (rank < KSEL) { ind[rank] = i; u[rank] = gi; }
}

// ---------------------------------------------------------------------------
// 2) per-selected-index prep: vals, w=sqrt(vals), w*z, KL reduction
// ---------------------------------------------------------------------------
__global__ void prep_kernel(const float* __restrict__ z,
                            const float* __restrict__ sigma2,
                            const float* __restrict__ p,
                            const float* __restrict__ pq,
                            const int* __restrict__ Nptr,
                            float* __restrict__ ws) {
  __shared__ float red[KSEL];
  const int t = threadIdx.x;
  const int* ind = (const int*)(ws + WS_IND);
  const int idx = ind[t];
  const float uv = ws[WS_U + t];
  const float mask = (uv > THRESHV) ? 1.0f : 0.0f;
  const float Nf = (float)(*Nptr);
  const float vals = Nf * sigma2[idx] * uv * mask;
  const float wt = sqrtf(vals);
  ws[WS_W + t]  = wt;
  ws[WS_BZ + t] = wt * z[idx];
  const float x1 = p[idx], x2 = pq[idx];
  const float kl = mask * (x2 * (logf(x2) - logf(x1)) +
                           (1.0f - x2) * (logf(1.0f - x2) - logf(1.0f - x1)));
  red[t] = kl;
  __syncthreads();
  for (int s = KSEL / 2; s > 0; s >>= 1) {
    if (t < s) red[t] += red[t + s];
    __syncthreads();
  }
  if (t == 0) ws[WS_SCAL + 0] = red[0];
}

// ---------------------------------------------------------------------------
// 3) z . S (8192 dot)
// ---------------------------------------------------------------------------
__global__ void zs_kernel(const float* __restrict__ z,
                          const float* __restrict__ S,
                          float* __restrict__ ws) {
  __shared__ float red[1024];
  const int t = threadIdx.x;
  const float4* z4 = (const float4*)z;
  const float4* S4 = (const float4*)S;
  float acc = 0.0f;
  for (int k = t; k < MDIM / 4; k += 1024) {
    float4 a = z4[k], b = S4[k];
    acc += a.x * b.x + a.y * b.y + a.z * b.z + a.w * b.w;
  }
  red[t] = acc;
  __syncthreads();
  for (int s = 512; s > 0; s >>= 1) {
    if (t < s) red[t] += red[t + s];
    __syncthreads();
  }
  if (t == 0) ws[WS_SCAL + 1] = red[0];
}

// ---------------------------------------------------------------------------
// 4) M = I + W * Rkk * W  (SPD; same det / quad form as sigma by Sylvester)
// ---------------------------------------------------------------------------
__global__ void msym_kernel(const float* __restrict__ R, float* __restrict__ ws) {
  const int i = blockIdx.x, j = threadIdx.x;
  const int* ind = (const int*)(ws + WS_IND);
  const float r = R[(size_t)ind[i] * MDIM + ind[j]];
  const float m = ws[WS_W + i] * ws[WS_W + j] * r + ((i == j) ? 1.0f : 0.0f);
  ws[WS_MSYM + i * KSEL + j] = m;
}

// ---------------------------------------------------------------------------
// 5) bv = W * (Rk @ S): one block per selected row, vectorized float4 loads.
//    8 MB streamed from HBM -> bandwidth-bound leg of the whole problem.
// ---------------------------------------------------------------------------
__global__ void gemv_kernel(const float* __restrict__ R,
                            const float* __restrict__ S,
                            float* __restrict__ ws) {
  __shared__ float red[256];
  const int r = blockIdx.x, t = threadIdx.x;
  const int* ind = (const int*)(ws + WS_IND);
  const float4* Rrow = (const float4*)(R + (size_t)ind[r] * MDIM);
  const float4* S4 = (const float4*)S;
  float acc = 0.0f;
  for (int k = t; k < MDIM / 4; k += 256) {
    float4 a = Rrow[k], b = S4[k];
    acc += a.x * b.x + a.y * b.y + a.z * b.z + a.w * b.w;
  }
  red[t] = acc;
  __syncthreads();
  for (int s = 128; s > 0; s >>= 1) {
    if (t < s) red[t] += red[t + s];
    __syncthreads();
  }
  if (t == 0) ws[WS_BV + r] = ws[WS_W + r] * red[0];
}

// ---------------------------------------------------------------------------
// 6) Blocked Cholesky (NB=16) of M in-place -> L (lower), plus logdet.
//    - diag 16x16 factored by a single wave32 with shuffles (no barriers)
//    - panel solve: one thread per trailing row
//    - trailing update: V_WMMA_F32_16X16X4_F32, 4 chained K=4 per tile
// ---------------------------------------------------------------------------
__global__ void __launch_bounds__(512) chol_kernel(float* __restrict__ ws) {
  float* Ms = ws + WS_MSYM;
  __shared__ float Lb[16 * 16];
  __shared__ float red[512];
  const int tid = threadIdx.x;
  const int lane = tid & 31;
  const int wave = tid >> 5;

  for (int pstep = 0; pstep < 16; ++pstep) {
    const int kp = pstep * 16;

    // ---- Phase A: factor diagonal block (wave 0, shuffle-based, no barriers)
    if (wave == 0) {
      const int row = lane & 15;  // lanes 16..31 mirror 0..15 (benign dup)
      float a[16];
#pragma unroll
      for (int k = 0; k < 16; ++k) a[k] = Ms[(kp + row) * KSEL + kp + k];
#pragma unroll
      for (int k = 0; k < 16; ++k) {
        const float akk = __shfl(a[k], k);
        const float d = sqrtf(akk);
        const float inv = 1.0f / d;
        if (row == k) a[k] = d;
        else if (row > k) a[k] *= inv;
#pragma unroll
        for (int j = k + 1; j < 16; ++j) {
          const float ljk = __shfl(a[k], j);   // L[j][k], already scaled
          if (row > k) a[j] -= a[k] * ljk;
        }
      }
#pragma unroll
      for (int k = 0; k < 16; ++k) {
        Ms[(kp + row) * KSEL + kp + k] = a[k];
        Lb[row * 16 + k] = a[k];
      }
    }
    __threadfence_block();
    __syncthreads();

    // ---- Phase B: panel solve  L_panel = A_panel * Lb^{-T}, one thread/row
    const int rstart = kp + 16;
    const int nrows = KSEL - rstart;
    if (tid < nrows) {
      const int i = rstart + tid;
      float y[16];
#pragma unroll
      for (int k = 0; k < 16; ++k) y[k] = Ms[i * KSEL + kp + k];
#pragma unroll
      for (int k = 0; k < 16; ++k) {
        float acc = y[k];
#pragma unroll
        for (int t2 = 0; t2 < 16; ++t2)
          if (t2 < k) acc -= y[t2] * Lb[k * 16 + t2];
        y[k] = acc / Lb[k * 16 + k];
      }
#pragma unroll
      for (int k = 0; k < 16; ++k) Ms[i * KSEL + kp + k] = y[k];
    }
    __threadfence_block();
    __syncthreads();

    // ---- Phase C: trailing update C -= P P^T via WMMA f32 16x16x4 chains
    const int rblk = 15 - pstep;
    const int ntiles = rblk * rblk;  // full square keeps matrix symmetric
    for (int tIdx = wave; tIdx < ntiles; tIdx += 16) {
      const int ib = pstep + 1 + tIdx / rblk;
      const int jb = pstep + 1 + tIdx % rblk;
      const int mn = lane & 15;                  // A row m / B col n / C col n
      const int khalf = (lane < 16) ? 0 : 2;     // K pair per lane half
      const int moff = (lane < 16) ? 0 : 8;      // C/D row offset per half

      v8f c;
#pragma unroll
      for (int rr = 0; rr < 8; ++rr)
        c[rr] = Ms[(ib * 16 + rr + moff) * KSEL + jb * 16 + mn];

#pragma unroll
      for (int cc = 0; cc < 4; ++cc) {
        v2f aop, bop;
        // A = -P_ib (16x4 slice), B = P_jb^T (4x16 slice)
        aop.x = -Ms[(ib * 16 + mn) * KSEL + kp + cc * 4 + khalf + 0];
        aop.y = -Ms[(ib * 16 + mn) * KSEL + kp + cc * 4 + khalf + 1];
        bop.x =  Ms[(jb * 16 + mn) * KSEL + kp + cc * 4 + khalf + 0];
        bop.y =  Ms[(jb * 16 + mn) * KSEL + kp + cc * 4 + khalf + 1];
        c = __builtin_amdgcn_wmma_f32_16x16x4_f32(
                false, aop, false, bop, (short)0, c, false, false);
      }
#pragma unroll
      for (int rr = 0; rr < 8; ++rr)
        Ms[(ib * 16 + rr + moff) * KSEL + jb * 16 + mn] = c[rr];
    }
    __threadfence_block();
    __syncthreads();
  }

  // ---- logdet = 2 * sum(log(diag(L)))
  float part = 0.0f;
  for (int i = tid; i < KSEL; i += 512) part += logf(Ms[i * KSEL + i]);
  red[tid] = part;
  __syncthreads();
  for (int s = 256; s > 0; s >>= 1) {
    if (tid < s) red[tid] += red[tid + s];
    __syncthreads();
  }
  if (tid == 0) ws[WS_SCAL + 2] = 2.0f * red[0];
}

// ---------------------------------------------------------------------------
// 7) dual forward solve L y = bz, L y2 = bv in one wave (shuffle broadcast,
//    zero barriers); t = sum yz_j * yv_j = bz^T M^{-1} bv; finalize scalar.
// ---------------------------------------------------------------------------
__global__ void solve_kernel(float* __restrict__ ws, float* __restrict__ out) {
  const int lane = threadIdx.x;  // 32 threads
  const float* Ms = ws + WS_MSYM;
  float rz[8], rv[8];
#pragma unroll
  for (int q = 0; q < 8; ++q) {
    rz[q] = ws[WS_BZ + q * 32 + lane];
    rv[q] = ws[WS_BV + q * 32 + lane];
  }
  float tpart = 0.0f;
#pragma unroll
  for (int qj = 0; qj < 8; ++qj) {
    for (int lj = 0; lj < 32; ++lj) {
      const int j = qj * 32 + lj;
      const float Ljj = Ms[j * KSEL + j];
      const float yz = __shfl(rz[qj], lj) / Ljj;
      const float yv = __shfl(rv[qj], lj) / Ljj;
      if (lane == lj) tpart += yz * yv;
#pragma unroll
      for (int q = 0; q < 8; ++q) {
        const int i = q * 32 + lane;
        if (i > j) {
          const float Lij = Ms[(size_t)i * KSEL + j];
          rz[q] -= Lij * yz;
          rv[q] -= Lij * yv;
        }
      }
    }
  }
#pragma unroll
  for (int off = 16; off > 0; off >>= 1) tpart += __shfl_xor(tpart, off);
  if (lane == 0) {
    const float kl = ws[WS_SCAL + 0];
    const float zs = ws[WS_SCAL + 1];
    const float logdet = ws[WS_SCAL + 2];
    // result = logdet/2 + (z.S - t)/2 + kl
    out[0] = 0.5f * logdet + 0.5f * (zs - tpart) + kl;
  }
}

extern "C" void kernel_launch(void* const* d_in, const int* in_sizes, int n_in,
                              void* d_out, int out_size, void* d_ws, size_t ws_size,
                              hipStream_t stream) {
  const float* R      = (const float*)d_in[0];
  const float* z      = (const float*)d_in[1];
  const float* sigma2 = (const float*)d_in[2];
  const float* p      = (const float*)d_in[3];
  const float* pq     = (const float*)d_in[4];
  const float* gamma  = (const float*)d_in[5];
  const float* S      = (const float*)d_in[6];
  const int*   N      = (const int*)d_in[7];
  float* ws  = (float*)d_ws;
  float* out = (float*)d_out;

  topk_kernel <<<32, 256, 0, stream>>>(gamma, (int*)(ws + WS_IND), ws + WS_U);
  prep_kernel <<<1, 256, 0, stream>>>(z, sigma2, p, pq, N, ws);
  zs_kernel   <<<1, 1024, 0, stream>>>(z, S, ws);
  msym_kernel <<<256, 256, 0, stream>>>(R, ws);
  gemv_kernel <<<256, 256, 0, stream>>>(R, S, ws);
  chol_kernel <<<1, 512, 0, stream>>>(ws);
  solve_kernel<<<1, 32, 0, stream>>>(ws, out);
}